// Net_9225589752152
// MI455X (gfx1250) — compile-verified
//
#include <hip/hip_runtime.h>
#include <hip/hip_bf16.h>
#include <math.h>

typedef __attribute__((ext_vector_type(2))) float v2f;
typedef __attribute__((ext_vector_type(8))) float v8f;

#define NEG_INF_BITS 0xFF800000u

__device__ __forceinline__ float finite_or_zero(float v) {
    // segment_max inits at -inf; clamp non-finite (inf/nan) to 0 like reference
    return ((__float_as_uint(v) & 0x7f800000u) != 0x7f800000u) ? v : 0.0f;
}

__device__ __forceinline__ void atomicMaxF(float* addr, float v) {
    // IEEE ordering trick: ints order like floats when >=0; for negatives the
    // unsigned ordering is reversed. Mixed updates commute correctly.
    if (v >= 0.0f) atomicMax((int*)addr, __float_as_int(v));
    else           atomicMin((unsigned int*)addr, __float_as_uint(v));
}

// ---------------------------------------------------------------- init
__global__ __launch_bounds__(256) void fill_f32(float* __restrict__ p, float v, int n) {
    int i = blockIdx.x * 256 + threadIdx.x;
    if (i < n) p[i] = v;
}

// ---------------------------------------------------------------- layer1 edge scatter
// per edge: gather x[src] (2 f32), weighted atomic-add + atomic-max into dst aggs
__global__ __launch_bounds__(256) void edge_scatter1(
    const int* __restrict__ src, const int* __restrict__ dst,
    const float* __restrict__ norm, const float* __restrict__ x,
    float* __restrict__ asum, float* __restrict__ amax, int e)
{
    int i = blockIdx.x * 256 + threadIdx.x;
    if (i >= e) return;
    int s = src[i], d = dst[i];
    float w  = norm[i];
    float x0 = x[s * 2 + 0];
    float x1 = x[s * 2 + 1];
    atomicAdd(&asum[d * 2 + 0], x0 * w);
    atomicAdd(&asum[d * 2 + 1], x1 * w);
    atomicMaxF(&amax[d * 2 + 0], x0);
    atomicMaxF(&amax[d * 2 + 1], x1);
}

// ---------------------------------------------------------------- layer2 edge scatter
// 8 threads per edge, each owning a float4 feature chunk of h[src]
__global__ __launch_bounds__(256) void edge_scatter2(
    const int* __restrict__ src, const int* __restrict__ dst,
    const float* __restrict__ norm, const float* __restrict__ h,
    float* __restrict__ asum, float* __restrict__ amax, int e)
{
    int i = blockIdx.x * 256 + threadIdx.x;
    int eidx = i >> 3;
    if (eidx >= e) return;
    int ch = (i & 7) * 4;
    int s = src[eidx], d = dst[eidx];
    float w = norm[eidx];
    const float4 hv = *(const float4*)(h + s * 32 + ch);
    float* as = asum + d * 32 + ch;
    float* am = amax + d * 32 + ch;
    atomicAdd(as + 0, hv.x * w);
    atomicAdd(as + 1, hv.y * w);
    atomicAdd(as + 2, hv.z * w);
    atomicAdd(as + 3, hv.w * w);
    atomicMaxF(am + 0, hv.x);
    atomicMaxF(am + 1, hv.y);
    atomicMaxF(am + 2, hv.z);
    atomicMaxF(am + 3, hv.w);
}

// ---------------------------------------------------------------- layer1 dense (WMMA f32)
// h[n,32] = relu( [asum|amaxClamped|x|pad](K=8) @ Wc[8,32] + bias )
// Wc block structure encodes lin_l(agg)+lin_r(x) for mean(24) and max(8) branches.
__global__ __launch_bounds__(256) void l1_dense(
    const float* __restrict__ x, const float* __restrict__ asum, const float* __restrict__ amax,
    const float* __restrict__ w1m_l, const float* __restrict__ b1m, const float* __restrict__ w1m_r,
    const float* __restrict__ w1x_l, const float* __restrict__ b1x, const float* __restrict__ w1x_r,
    float* __restrict__ h, int n)
{
    __shared__ float Wc[8][32];
    __shared__ float Bc[32];
    int t = threadIdx.x;
    {   // 8*32 = 256 weight entries, one per thread
        int k = t >> 5, nn = t & 31;
        float v = 0.0f;
        if (nn < 24) {                       // weighted-mean branch, K: [aggsum(0:2) | - | x(4:6)]
            if (k < 2)               v = w1m_l[nn * 2 + k];
            else if (k >= 4 && k < 6) v = w1m_r[nn * 2 + (k - 4)];
        } else {                             // max branch, K: [- | aggmax(2:4) | x(4:6)]
            int nx = nn - 24;
            if (k >= 2 && k < 4)      v = w1x_l[nx * 2 + (k - 2)];
            else if (k >= 4 && k < 6) v = w1x_r[nx * 2 + (k - 4)];
        }
        Wc[k][nn] = v;
        if (t < 32) Bc[t] = (t < 24) ? b1m[t] : b1x[t - 24];
    }
    __syncthreads();

    int wave = t >> 5, lane = t & 31;
    int m0 = (blockIdx.x * 8 + wave) * 16;
    if (m0 >= n) return;
    int row = m0 + (lane & 15);
    bool valid = row < n;
    int khalf = (lane >> 4) << 1;            // lanes 16-31 carry K+2,K+3
    int col = lane & 15;

    float f[8];
    if (valid) {
        f[0] = asum[row * 2 + 0];
        f[1] = asum[row * 2 + 1];
        f[2] = finite_or_zero(amax[row * 2 + 0]);
        f[3] = finite_or_zero(amax[row * 2 + 1]);
        f[4] = x[row * 2 + 0];
        f[5] = x[row * 2 + 1];
    } else {
        f[0] = f[1] = f[2] = f[3] = f[4] = f[5] = 0.0f;
    }
    f[6] = 0.0f; f[7] = 0.0f;

    v8f c0 = {0, 0, 0, 0, 0, 0, 0, 0};
    v8f c1 = {0, 0, 0, 0, 0, 0, 0, 0};
#pragma unroll
    for (int k0 = 0; k0 < 8; k0 += 4) {
        v2f a;  a.x  = f[k0 + khalf];            a.y  = f[k0 + khalf + 1];
        v2f b0; b0.x = Wc[k0 + khalf][col];      b0.y = Wc[k0 + khalf + 1][col];
        v2f b1; b1.x = Wc[k0 + khalf][col + 16]; b1.y = Wc[k0 + khalf + 1][col + 16];
        c0 = __builtin_amdgcn_wmma_f32_16x16x4_f32(false, a, false, b0, (short)0, c0, false, false);
        c1 = __builtin_amdgcn_wmma_f32_16x16x4_f32(false, a, false, b1, (short)0, c1, false, false);
    }

    int mhi = (lane >> 4) * 8;               // D: lanes 16-31 hold M=8..15
#pragma unroll
    for (int i = 0; i < 8; ++i) {
        int r = m0 + mhi + i;
        if (r < n) {
            float v0 = c0[i] + Bc[col];
            float v1 = c1[i] + Bc[col + 16];
            h[r * 32 + col]      = v0 > 0.0f ? v0 : 0.0f;
            h[r * 32 + col + 16] = v1 > 0.0f ? v1 : 0.0f;
        }
    }
}

// ---------------------------------------------------------------- layer2 dense (WMMA f32)
// h2[n,16] = relu( [asum2(32)|amax2Clamped(32)|h(32)](K=96) @ Wc[96,16] + bias )
__global__ __launch_bounds__(256) void l2_dense(
    const float* __restrict__ h, const float* __restrict__ asum, const float* __restrict__ amax,
    const float* __restrict__ w2m_l, const float* __restrict__ b2m, const float* __restrict__ w2m_r,
    const float* __restrict__ w2x_l, const float* __restrict__ b2x, const float* __restrict__ w2x_r,
    float* __restrict__ h2, int n)
{
    __shared__ float Wc[96][16];
    __shared__ float Bc[16];
    int t = threadIdx.x;
    for (int i = t; i < 96 * 16; i += 256) {
        int k = i >> 4, nn = i & 15;
        float v = 0.0f;
        if (nn < 12) {                       // mean branch: [aggsum | 0 | h]
            if (k < 32)       v = w2m_l[nn * 32 + k];
            else if (k >= 64) v = w2m_r[nn * 32 + (k - 64)];
        } else {                             // max branch: [0 | aggmax | h]
            int nx = nn - 12;
            if (k >= 32 && k < 64) v = w2x_l[nx * 32 + (k - 32)];
            else if (k >= 64)      v = w2x_r[nx * 32 + (k - 64)];
        }
        Wc[k][nn] = v;
    }
    if (t < 16) Bc[t] = (t < 12) ? b2m[t] : b2x[t - 12];
    __syncthreads();

    int wave = t >> 5, lane = t & 31;
    int m0 = (blockIdx.x * 8 + wave) * 16;
    if (m0 >= n) return;
    int row = m0 + (lane & 15);
    bool valid = row < n;
    int khalf = (lane >> 4) << 1;
    int col = lane & 15;

    v8f c = {0, 0, 0, 0, 0, 0, 0, 0};
#pragma unroll
    for (int k0 = 0; k0 < 96; k0 += 4) {
        int kk = k0 + khalf;
        float ax = 0.0f, ay = 0.0f;
        if (valid) {
            if (k0 < 32) {
                ax = asum[row * 32 + kk];
                ay = asum[row * 32 + kk + 1];
            } else if (k0 < 64) {
                ax = finite_or_zero(amax[row * 32 + kk - 32]);
                ay = finite_or_zero(amax[row * 32 + kk - 31]);
            } else {
                ax = h[row * 32 + kk - 64];
                ay = h[row * 32 + kk - 63];
            }
        }
        v2f a; a.x = ax;          a.y = ay;
        v2f b; b.x = Wc[kk][col]; b.y = Wc[kk + 1][col];
        c = __builtin_amdgcn_wmma_f32_16x16x4_f32(false, a, false, b, (short)0, c, false, false);
    }

    int mhi = (lane >> 4) * 8;
#pragma unroll
    for (int i = 0; i < 8; ++i) {
        int r = m0 + mhi + i;
        if (r < n) {
            float v = c[i] + Bc[col];
            h2[r * 16 + col] = v > 0.0f ? v : 0.0f;
        }
    }
}

// ---------------------------------------------------------------- MLP head 16->8->5->1
__global__ __launch_bounds__(256) void head_mlp(
    const float* __restrict__ h2,
    const float* __restrict__ w3, const float* __restrict__ b3,
    const float* __restrict__ w4, const float* __restrict__ b4,
    const float* __restrict__ w5, const float* __restrict__ b5,
    float* __restrict__ out, int n)
{
    int i = blockIdx.x * 256 + threadIdx.x;
    if (i >= n) return;
    float v[16];
#pragma unroll
    for (int j = 0; j < 16; ++j) v[j] = h2[i * 16 + j];
    float a1[8];
#pragma unroll
    for (int o = 0; o < 8; ++o) {
        float s = b3[o];
#pragma unroll
        for (int j = 0; j < 16; ++j) s += w3[o * 16 + j] * v[j];
        a1[o] = s > 0.0f ? s : 0.0f;
    }
    float a2[5];
#pragma unroll
    for (int o = 0; o < 5; ++o) {
        float s = b4[o];
#pragma unroll
        for (int j = 0; j < 8; ++j) s += w4[o * 8 + j] * a1[j];
        a2[o] = s > 0.0f ? s : 0.0f;
    }
    float s = b5[0];
#pragma unroll
    for (int j = 0; j < 5; ++j) s += w5[j] * a2[j];
    out[i] = s;
}

// ---------------------------------------------------------------- launch
extern "C" void kernel_launch(void* const* d_in, const int* in_sizes, int n_in,
                              void* d_out, int out_size, void* d_ws, size_t ws_size,
                              hipStream_t stream) {
    const float* x     = (const float*)d_in[0];
    const int*   eidx  = (const int*)d_in[1];
    const float* norm  = (const float*)d_in[2];
    const float* w1m_l = (const float*)d_in[3];
    const float* b1m   = (const float*)d_in[4];
    const float* w1m_r = (const float*)d_in[5];
    const float* w1x_l = (const float*)d_in[6];
    const float* b1x   = (const float*)d_in[7];
    const float* w1x_r = (const float*)d_in[8];
    const float* w2m_l = (const float*)d_in[9];
    const float* b2m   = (const float*)d_in[10];
    const float* w2m_r = (const float*)d_in[11];
    const float* w2x_l = (const float*)d_in[12];
    const float* b2x   = (const float*)d_in[13];
    const float* w2x_r = (const float*)d_in[14];
    const float* w3    = (const float*)d_in[15];
    const float* b3    = (const float*)d_in[16];
    const float* w4    = (const float*)d_in[17];
    const float* b4    = (const float*)d_in[18];
    const float* w5    = (const float*)d_in[19];
    const float* b5    = (const float*)d_in[20];
    float* out = (float*)d_out;

    const int n = in_sizes[0] / 2;          // 100000
    const int e = in_sizes[2];              // 3200000
    const int* src = eidx;                  // edge_index[0]
    const int* dst = eidx + e;              // edge_index[1]

    // workspace layout (f32 elements), 256B-aligned sections
    float* ws = (float*)d_ws;
    size_t off = 0;
    auto take = [&](size_t elems) { float* p = ws + off; off += (elems + 63) & ~size_t(63); return p; };
    float* asum1 = take((size_t)n * 2);
    float* amax1 = take((size_t)n * 2);
    float* h     = take((size_t)n * 32);
    float* asum2 = take((size_t)n * 32);
    float* amax2 = take((size_t)n * 32);
    float* h2    = take((size_t)n * 16);
    (void)ws_size;

    const float NEG_INF = __builtin_bit_cast(float, NEG_INF_BITS);
    auto grid1 = [](int cnt) { return dim3((cnt + 255) / 256); };

    // 1. init aggregates
    fill_f32<<<grid1(n * 2),  256, 0, stream>>>(asum1, 0.0f,    n * 2);
    fill_f32<<<grid1(n * 2),  256, 0, stream>>>(amax1, NEG_INF, n * 2);
    fill_f32<<<grid1(n * 32), 256, 0, stream>>>(asum2, 0.0f,    n * 32);
    fill_f32<<<grid1(n * 32), 256, 0, stream>>>(amax2, NEG_INF, n * 32);

    // 2. layer-1 edge scatter
    edge_scatter1<<<grid1(e), 256, 0, stream>>>(src, dst, norm, x, asum1, amax1, e);

    // 3. layer-1 dense (WMMA f32): h[n,32]
    int tiles_blocks = (n + 128 - 1) / 128;   // 8 waves x 16 nodes per block
    l1_dense<<<tiles_blocks, 256, 0, stream>>>(x, asum1, amax1,
                                               w1m_l, b1m, w1m_r, w1x_l, b1x, w1x_r, h, n);

    // 4. layer-2 edge scatter (8 threads/edge)
    edge_scatter2<<<grid1(e * 8), 256, 0, stream>>>(src, dst, norm, h, asum2, amax2, e);

    // 5. layer-2 dense (WMMA f32): h2[n,16]
    l2_dense<<<tiles_blocks, 256, 0, stream>>>(h, asum2, amax2,
                                               w2m_l, b2m, w2m_r, w2x_l, b2x, w2x_r, h2, n);

    // 6. MLP head
    head_mlp<<<grid1(n), 256, 0, stream>>>(h2, w3, b3, w4, b4, w5, b5, out, n);
}